// Embedding_82463372083922
// MI455X (gfx1250) — compile-verified
//
#include <hip/hip_runtime.h>
#include <stdint.h>

// Embedding gather: out[row, :] = weight[token_ids[row], :]
// token_ids: int32[8192], weight: float32[32000, 1024], out: float32[8192, 1024]
//
// Memory-bound row gather (~64 MiB traffic -> ~2.9 us floor at 23.3 TB/s).
// CDNA5 async global<->LDS DMA path (ASYNCcnt): row data never occupies data
// VGPRs. 4 rows per block staged in LDS; each wave keeps 4 async loads (2 KiB)
// in flight before a single s_wait_asynccnt, then streams 4 async stores
// tagged TH_STORE_NT so the write-once 32 MiB output doesn't evict the weight
// table (128 MiB, resident in the 192 MB L2).
//
// Fast path (full blocks): all 4 token-index s_loads cluster under one
// s_wait_kmcnt, no per-row branches. Guarded tail path handles any n_tokens.

#define EMBED 1024   // floats per row (matches reference)
#define BLOCK 256    // 256 lanes * 16 B = 4096 B = one row per async op
#define ROWS  4      // rows staged per block (16 KiB LDS of 320 KiB/WGP)

__global__ __launch_bounds__(BLOCK) void embed_gather_async_kernel(
    const int* __restrict__ token_ids,
    const float* __restrict__ weight,
    float* __restrict__ out,
    int n_tokens)
{
    __shared__ float smem[ROWS][EMBED];   // 4 x 4 KiB staging buffers

    const int base = blockIdx.x * ROWS;
    const int t    = threadIdx.x;         // lane's 16-byte chunk within a row

    if (base + ROWS <= n_tokens) {
        // ---------------- Fast path: full block, no guards ----------------
        // Gather all indices first so the scalar loads issue as one clause
        // with a single kmcnt wait instead of serializing per row.
        int idx[ROWS];
#pragma unroll
        for (int b = 0; b < ROWS; ++b)
            idx[b] = token_ids[base + b];            // clustered s_load_b32

        // Issue ROWS async DMA loads (global -> LDS) back-to-back.
#pragma unroll
        for (int b = 0; b < ROWS; ++b) {
            const uint64_t src =
                (uint64_t)(weight + (size_t)idx[b] * EMBED + (size_t)t * 4);
            const uint32_t lds = (uint32_t)(uintptr_t)(&smem[b][t * 4]);
            // 128b per lane, 512 B per wave, fully coalesced 4 KiB row.
            asm volatile("global_load_async_to_lds_b128 %0, %1, off"
                         :: "v"(lds), "v"(src)
                         : "memory");
        }

        // One wait for all of this wave's async loads. Each wave stores
        // exactly the LDS bytes its own loads wrote -> no barrier needed.
        asm volatile("s_wait_asynccnt 0x0" ::: "memory");

        // Issue ROWS async DMA stores (LDS -> global), non-temporal.
#pragma unroll
        for (int b = 0; b < ROWS; ++b) {
            const uint64_t dst =
                (uint64_t)(out + (size_t)(base + b) * EMBED + (size_t)t * 4);
            const uint32_t lds = (uint32_t)(uintptr_t)(&smem[b][t * 4]);
            asm volatile("global_store_async_from_lds_b128 %0, %1, off th:TH_STORE_NT"
                         :: "v"(dst), "v"(lds)
                         : "memory");
        }
    } else {
        // ---------------- Tail path: per-row uniform guards ----------------
#pragma unroll
        for (int b = 0; b < ROWS; ++b) {
            const int row = base + b;
            if (row < n_tokens) {
                const int idx = token_ids[row];
                const uint64_t src =
                    (uint64_t)(weight + (size_t)idx * EMBED + (size_t)t * 4);
                const uint32_t lds = (uint32_t)(uintptr_t)(&smem[b][t * 4]);
                asm volatile("global_load_async_to_lds_b128 %0, %1, off"
                             :: "v"(lds), "v"(src)
                             : "memory");
            }
        }

        asm volatile("s_wait_asynccnt 0x0" ::: "memory");

#pragma unroll
        for (int b = 0; b < ROWS; ++b) {
            const int row = base + b;
            if (row < n_tokens) {
                const uint64_t dst =
                    (uint64_t)(out + (size_t)row * EMBED + (size_t)t * 4);
                const uint32_t lds = (uint32_t)(uintptr_t)(&smem[b][t * 4]);
                asm volatile("global_store_async_from_lds_b128 %0, %1, off th:TH_STORE_NT"
                             :: "v"(dst), "v"(lds)
                             : "memory");
            }
        }
    }
    // S_ENDPGM performs an implicit wait-idle, draining outstanding async
    // stores before the wave retires.
}

extern "C" void kernel_launch(void* const* d_in, const int* in_sizes, int n_in,
                              void* d_out, int out_size, void* d_ws, size_t ws_size,
                              hipStream_t stream)
{
    const int*   token_ids = (const int*)d_in[0];     // int32[4*2048]
    const float* weight    = (const float*)d_in[1];   // f32[32000*1024]
    float*       out       = (float*)d_out;           // f32[8192*1024]

    const int n_tokens = in_sizes[0];                 // 8192
    const int n_blocks = (n_tokens + ROWS - 1) / ROWS;

    embed_gather_async_kernel<<<n_blocks, BLOCK, 0, stream>>>(
        token_ids, weight, out, n_tokens);
}